// FoldingNet_graph_51024211476655
// MI455X (gfx1250) — compile-verified
//
#include <hip/hip_runtime.h>
#include <hip/hip_bf16.h>
#include <math.h>

// ---------------------------------------------------------------------------
// FoldingNet forward for MI455X (gfx1250, wave32).
// GEMMs on the matrix pipe via V_WMMA_F32_16X16X4_F32 (fp32 end-to-end: the
// round(code) quantization makes the net discontinuous, so low precision is
// unsafe; the workload is HBM-bound anyway).
// Hot loop: 32x32 tile/wave (4 accumulators, 8 WMMA per 8-channel group),
// software double-buffered unconditional loads (clamped rows/cols, zero-
// padded K), compile-time column stride so B loads use immediate offsets.
// ---------------------------------------------------------------------------

typedef float v2f __attribute__((ext_vector_type(2)));
typedef float v8f __attribute__((ext_vector_type(8)));

static constexpr int BB = 16;     // batch
static constexpr int NN = 2048;   // points
static constexpr int KK = 16;     // neighbors
static constexpr int GRID = 45;
static constexpr int MM = GRID * GRID;  // 2025
static constexpr int DCH = 520;   // decoder input channels padded (514/515 -> 520)

#define WMMA_F32(A, Bv, C) \
    __builtin_amdgcn_wmma_f32_16x16x4_f32(false, (A), false, (Bv), (short)0, (C), false, false)

// ---------------------------------------------------------------------------
// fp32 WMMA GEMM:  Y[b,o,n] = sum_c W[o,c] * X[b,c,n] + bias[o]
//   X: [Bz, Cx, TN], zero-padded channels; Cx multiple of 8.
//   W: [O, Cx], zero-padded to Cx.
// One wave computes a 32(o) x 32(n) tile; K-loop loads are unconditional
// (rows/cols clamped, masked only at store) and double-buffered.
// ---------------------------------------------------------------------------
template <int TN>
__global__ void fn_gemm_wmma(const float* __restrict__ X,
                             const float* __restrict__ W,
                             const float* __restrict__ bias,
                             float* __restrict__ Y,
                             int Cx, int O, int doRelu)
{
    const int lane  = threadIdx.x;                       // 0..31
    const int nOt   = (O + 31) >> 5;                     // 32-row o-tiles
    const int otile = blockIdx.y * blockDim.y + threadIdx.y;
    if (otile >= nOt) return;                            // wave-uniform exit
    const int b  = blockIdx.z;
    const int n0 = blockIdx.x << 5;                      // 32 columns per wave
    const int o0 = otile << 5;

    const int l15 = lane & 15;
    const int kHi = (lane >> 4) << 1;                    // 0 or 2

    const int col0 = n0 + l15;
    const int col1 = col0 + 16;
    const int c0c  = min(col0, TN - 1);                  // clamp for loads
    const int c1c  = min(col1, TN - 1);
    const int row0 = min(o0 + l15, O - 1);
    const int row1 = min(o0 + 16 + l15, O - 1);

    const float* Xb  = X + (size_t)b * Cx * TN;
    const float* xp0 = Xb + c0c + kHi * TN;
    const float* xp1 = Xb + c1c + kHi * TN;
    const float* wp0 = W + (size_t)row0 * Cx + kHi;
    const float* wp1 = W + (size_t)row1 * Cx + kHi;

    v8f acc00 = {0.f,0.f,0.f,0.f,0.f,0.f,0.f,0.f};   // rows[0:16)  x col0
    v8f acc01 = acc00;                               // rows[0:16)  x col1
    v8f acc10 = acc00;                               // rows[16:32) x col0
    v8f acc11 = acc00;                               // rows[16:32) x col1

    // Per 8-channel group:
    //   aTH = A rows half T, K-sub H (v2f, b64 load)
    //   bCH = B col C, K-sub H  (two b32 loads at immediate offsets)
#define LOAD_GROUP(A00,A01,A10,A11, B00,B01,B10,B11)                         \
    A00 = *(const v2f*)(wp0);      A01 = *(const v2f*)(wp0 + 4);             \
    A10 = *(const v2f*)(wp1);      A11 = *(const v2f*)(wp1 + 4);             \
    B00.x = xp0[0];        B00.y = xp0[TN];                                  \
    B01.x = xp0[4 * TN];   B01.y = xp0[5 * TN];                              \
    B10.x = xp1[0];        B10.y = xp1[TN];                                  \
    B11.x = xp1[4 * TN];   B11.y = xp1[5 * TN];

#define DO_WMMAS(A00,A01,A10,A11, B00,B01,B10,B11)                           \
    acc00 = WMMA_F32(A00, B00, acc00);                                       \
    acc10 = WMMA_F32(A10, B00, acc10);                                       \
    acc01 = WMMA_F32(A00, B10, acc01);                                       \
    acc11 = WMMA_F32(A10, B10, acc11);                                       \
    acc00 = WMMA_F32(A01, B01, acc00);                                       \
    acc10 = WMMA_F32(A11, B01, acc10);                                       \
    acc01 = WMMA_F32(A01, B11, acc01);                                       \
    acc11 = WMMA_F32(A11, B11, acc11);

    v2f a00, a01, a10, a11, b00, b01, b10, b11;
    LOAD_GROUP(a00, a01, a10, a11, b00, b01, b10, b11)

    for (int kk = 8; kk < Cx; kk += 8) {
        wp0 += 8; wp1 += 8;
        xp0 += 8 * TN; xp1 += 8 * TN;
        // issue next group's loads BEFORE consuming current registers
        v2f na00, na01, na10, na11, nb00, nb01, nb10, nb11;
        LOAD_GROUP(na00, na01, na10, na11, nb00, nb01, nb10, nb11)
        DO_WMMAS(a00, a01, a10, a11, b00, b01, b10, b11)
        a00 = na00; a01 = na01; a10 = na10; a11 = na11;
        b00 = nb00; b01 = nb01; b10 = nb10; b11 = nb11;
    }
    DO_WMMAS(a00, a01, a10, a11, b00, b01, b10, b11)
#undef LOAD_GROUP
#undef DO_WMMAS

    // C/D layout: vgpr r -> (o = oBase + r, n = col),
    // oBase = tile_row_base + (lane>=16 ? 8 : 0)
    const int oB0 = o0 + ((lane >> 4) << 3);
    const int oB1 = oB0 + 16;
    float* Yb = Y + (size_t)b * O * TN;
    const bool ok0 = (col0 < TN), ok1 = (col1 < TN);
    #pragma unroll
    for (int r = 0; r < 8; ++r) {
        const int o = oB0 + r;
        if (o < O) {
            const float bo = bias[o];
            if (ok0) {
                float v = acc00[r] + bo;
                if (doRelu) v = fmaxf(v, 0.f);
                Yb[(size_t)o * TN + col0] = v;
            }
            if (ok1) {
                float v = acc01[r] + bo;
                if (doRelu) v = fmaxf(v, 0.f);
                Yb[(size_t)o * TN + col1] = v;
            }
        }
    }
    #pragma unroll
    for (int r = 0; r < 8; ++r) {
        const int o = oB1 + r;
        if (o < O) {
            const float bo = bias[o];
            if (ok0) {
                float v = acc10[r] + bo;
                if (doRelu) v = fmaxf(v, 0.f);
                Yb[(size_t)o * TN + col0] = v;
            }
            if (ok1) {
                float v = acc11[r] + bo;
                if (doRelu) v = fmaxf(v, 0.f);
                Yb[(size_t)o * TN + col1] = v;
            }
        }
    }
}

// ---------------------------------------------------------------------------
// Zero-pad a weight matrix [O,Cw] -> [O,Cp]
// ---------------------------------------------------------------------------
__global__ void fn_pad_w(const float* __restrict__ src, float* __restrict__ dst,
                         int O, int Cw, int Cp)
{
    int i = blockIdx.x * blockDim.x + threadIdx.x;
    if (i >= O * Cp) return;
    const int c = i % Cp, o = i / Cp;
    dst[i] = (c < Cw) ? src[(size_t)o * Cw + c] : 0.f;
}

// ---------------------------------------------------------------------------
// concat([x(3ch), Cov(9ch)], zero-pad to 16) -> [B,16,N]
// ---------------------------------------------------------------------------
__global__ void fn_concat16(const float* __restrict__ x,
                            const float* __restrict__ cov,
                            float* __restrict__ out)
{
    const int total = BB * 16 * NN;
    int i = blockIdx.x * blockDim.x + threadIdx.x;
    if (i >= total) return;
    const int n = i % NN;
    const int c = (i / NN) % 16;
    const int b = i / (NN * 16);
    float v = 0.f;
    if (c < 3)       v = x[((size_t)b * 3 + c) * NN + n];
    else if (c < 12) v = cov[((size_t)b * 9 + (c - 3)) * NN + n];
    out[i] = v;
}

// ---------------------------------------------------------------------------
// Batch-stat BN: per-channel mean/var over (B,N); folds to y = a*x + d.
// ---------------------------------------------------------------------------
__global__ void fn_bn_stats(const float* __restrict__ Y,
                            const float* __restrict__ g,
                            const float* __restrict__ bbias,
                            float* __restrict__ aOut, float* __restrict__ dOut,
                            int C, int Nn, int Bv)
{
    const int c = blockIdx.x;
    const int t = threadIdx.x;
    float s = 0.f, s2 = 0.f;
    for (int b = 0; b < Bv; ++b) {
        const float* p = Y + ((size_t)b * C + c) * Nn;
        for (int n = t; n < Nn; n += 256) { float v = p[n]; s += v; s2 += v * v; }
    }
    __shared__ float sh[256], sh2[256];
    sh[t] = s; sh2[t] = s2; __syncthreads();
    for (int st = 128; st > 0; st >>= 1) {
        if (t < st) { sh[t] += sh[t + st]; sh2[t] += sh2[t + st]; }
        __syncthreads();
    }
    if (t == 0) {
        const float cnt = (float)Bv * (float)Nn;
        const float m   = sh[0] / cnt;
        const float var = sh2[0] / cnt - m * m;
        const float a   = g[c] * rsqrtf(var + 1e-5f);
        aOut[c] = a;
        dOut[c] = bbias[c] - a * m;
    }
}

__global__ void fn_bn_apply_relu(float* __restrict__ Y,
                                 const float* __restrict__ a,
                                 const float* __restrict__ d,
                                 int C, int Nn, int total)
{
    int i = blockIdx.x * blockDim.x + threadIdx.x;
    if (i >= total) return;
    const int c = (i / Nn) % C;
    Y[i] = fmaxf(a[c] * Y[i] + d[c], 0.f);
}

// ---------------------------------------------------------------------------
// relu(max(max_k F[b,c,nb[b,n,k]], F[b,c,n])).
// One thread per (b, n, 16-channel group); int4-vectorized index loads.
// ---------------------------------------------------------------------------
__global__ void fn_graph_pool(const float* __restrict__ F,
                              const int* __restrict__ nb,
                              float* __restrict__ Out, int C)
{
    const int G = C >> 4;                       // channel groups of 16
    int i = blockIdx.x * blockDim.x + threadIdx.x;
    if (i >= BB * NN * G) return;
    const int n = i % NN;
    const int g = (i / NN) % G;
    const int b = i / (NN * G);

    int idx[KK];
    const int4* nbp = (const int4*)(nb + ((size_t)b * NN + n) * KK);
    #pragma unroll
    for (int q = 0; q < 4; ++q) {
        int4 v = nbp[q];
        idx[q * 4 + 0] = v.x; idx[q * 4 + 1] = v.y;
        idx[q * 4 + 2] = v.z; idx[q * 4 + 3] = v.w;
    }
    const float* Fb = F + (size_t)b * C * NN;
    float* Ob = Out + (size_t)b * C * NN;
    const int c0 = g << 4;
    for (int c = c0; c < c0 + 16; ++c) {
        const float* row = Fb + (size_t)c * NN;
        float m = row[n];
        #pragma unroll
        for (int k = 0; k < KK; ++k) m = fmaxf(m, row[idx[k]]);
        Ob[(size_t)c * NN + n] = fmaxf(m, 0.f);
    }
}

// ---------------------------------------------------------------------------
// h[b,c] = max_n (a[c]*Y[b,c,n] + d[c]), stored transposed [C][B].
// ---------------------------------------------------------------------------
__global__ void fn_maxpool_bn(const float* __restrict__ Y,
                              const float* __restrict__ a,
                              const float* __restrict__ d,
                              float* __restrict__ outT, int C, int Nn)
{
    const int bc = blockIdx.x;
    const int b = bc / C, c = bc % C;
    const int t = threadIdx.x;
    const float av = a[c], dv = d[c];
    const float* p = Y + ((size_t)b * C + c) * Nn;
    float m = -3.402823466e+38f;
    for (int n = t; n < Nn; n += 256) m = fmaxf(m, av * p[n] + dv);
    __shared__ float sh[256];
    sh[t] = m; __syncthreads();
    for (int st = 128; st > 0; st >>= 1) {
        if (t < st) sh[t] = fmaxf(sh[t], sh[t + st]);
        __syncthreads();
    }
    if (t == 0) outT[(size_t)c * BB + b] = sh[0];
}

// ---------------------------------------------------------------------------
// bn over batch dim for FC activations stored [R][B], then relu, in place.
// ---------------------------------------------------------------------------
__global__ void fn_bn_fc_relu(float* __restrict__ X,
                              const float* __restrict__ g,
                              const float* __restrict__ bbias, int R)
{
    int r = blockIdx.x * blockDim.x + threadIdx.x;
    if (r >= R) return;
    float* p = X + (size_t)r * BB;
    float s = 0.f, s2 = 0.f;
    for (int i = 0; i < BB; ++i) { float v = p[i]; s += v; s2 += v * v; }
    const float m   = s / BB;
    const float var = s2 / BB - m * m;
    const float a   = g[r] * rsqrtf(var + 1e-5f);
    const float dd  = bbias[r] - a * m;
    for (int i = 0; i < BB; ++i) p[i] = fmaxf(a * p[i] + dd, 0.f);
}

// codeT [512][B] -> rint in place (round-half-even == jnp.round), and emit
// code output [B][512].
__global__ void fn_round_code(float* __restrict__ codeT,
                              float* __restrict__ codeOut)
{
    int i = blockIdx.x * blockDim.x + threadIdx.x;
    if (i >= 512 * BB) return;
    const float v = rintf(codeT[i]);
    codeT[i] = v;
    const int r = i / BB, b = i % BB;
    codeOut[(size_t)b * 512 + r] = v;
}

// Decoder input D [B,DCH,M]: ch<512 = codeT[c][b]; 512/513 = grid xy; rest 0.
__global__ void fn_build_dec_in(const float* __restrict__ codeT,
                                float* __restrict__ D)
{
    const size_t total = (size_t)BB * DCH * MM;
    size_t i = (size_t)blockIdx.x * blockDim.x + threadIdx.x;
    if (i >= total) return;
    const int m = (int)(i % MM);
    const int c = (int)((i / MM) % DCH);
    const int b = (int)(i / ((size_t)MM * DCH));
    float v;
    if (c < 512)       v = codeT[(size_t)c * BB + b];
    else if (c == 512) v = -0.3f + 0.6f * (float)(m % GRID) / (float)(GRID - 1);
    else if (c == 513) v = -0.3f + 0.6f * (float)(m / GRID) / (float)(GRID - 1);
    else               v = 0.f;
    D[i] = v;
}

// Overwrite D channels 512..514 with p1 [B,3,M] (515..519 stay 0).
__global__ void fn_patch_p1(const float* __restrict__ p1,
                            float* __restrict__ D)
{
    int i = blockIdx.x * blockDim.x + threadIdx.x;
    if (i >= BB * 3 * MM) return;
    const int m  = i % MM;
    const int ch = (i / MM) % 3;
    const int b  = i / (MM * 3);
    D[((size_t)b * DCH + 512 + ch) * MM + m] = p1[i];
}

// ---------------------------------------------------------------------------
extern "C" void kernel_launch(void* const* d_in, const int* in_sizes, int n_in,
                              void* d_out, int out_size, void* d_ws, size_t ws_size,
                              hipStream_t stream)
{
    (void)in_sizes; (void)n_in; (void)out_size; (void)ws_size;

    const float* x    = (const float*)d_in[0];   // [16,3,2048]
    const float* Cov  = (const float*)d_in[1];   // [16,9,2048]
    const int*   nb   = (const int*)  d_in[2];   // [16,2048,16]
    const float* c1w = (const float*)d_in[3],  *c1b = (const float*)d_in[4];
    const float* c2w = (const float*)d_in[5],  *c2b = (const float*)d_in[6];
    const float* c3w = (const float*)d_in[7],  *c3b = (const float*)d_in[8];
    const float* c4w = (const float*)d_in[9],  *c4b = (const float*)d_in[10];
    const float* c5w = (const float*)d_in[11], *c5b = (const float*)d_in[12];
    const float* f1w = (const float*)d_in[13], *f1b = (const float*)d_in[14];
    const float* f2w = (const float*)d_in[15], *f2b = (const float*)d_in[16];
    const float* g1w = (const float*)d_in[17], *g1b = (const float*)d_in[18];
    const float* g2w = (const float*)d_in[19], *g2b = (const float*)d_in[20];
    const float* g3w = (const float*)d_in[21], *g3b = (const float*)d_in[22];
    const float* h1w = (const float*)d_in[23], *h1b = (const float*)d_in[24];
    const float* h2w = (const float*)d_in[25], *h2b = (const float*)d_in[26];
    const float* h3w = (const float*)d_in[27], *h3b = (const float*)d_in[28];
    const float* bn1g = (const float*)d_in[29], *bn1b = (const float*)d_in[30];
    const float* bn2g = (const float*)d_in[31], *bn2b = (const float*)d_in[32];
    const float* bn3g = (const float*)d_in[33], *bn3b = (const float*)d_in[34];
    const float* bn4g = (const float*)d_in[35], *bn4b = (const float*)d_in[36];
    const float* bn5g = (const float*)d_in[37], *bn5b = (const float*)d_in[38];
    const float* bn6g = (const float*)d_in[39], *bn6b = (const float*)d_in[40];

    // outputs: out [16,3,2025], p1 [16,3,2025], code [16,512]
    float* outP  = (float*)d_out;
    float* p1P   = outP + (size_t)BB * 3 * MM;       // 97200
    float* codeP = p1P  + (size_t)BB * 3 * MM;       // 194400

    // --- workspace partition (fp32) ---
    float* ws = (float*)d_ws;
    size_t off = 0;
    auto take = [&](size_t n) { float* p = ws + off; off += n; return p; };
    float* cat16  = take((size_t)BB * 16 * NN);      //   2 MB
    float* t64a   = take((size_t)BB * 64 * NN);      //   8 MB
    float* t64b   = take((size_t)BB * 64 * NN);      //   8 MB
    float* t128a  = take((size_t)BB * 128 * NN);     //  16 MB
    float* t128b  = take((size_t)BB * 128 * NN);     //  16 MB
    float* big    = take((size_t)BB * 1024 * NN);    // 134 MB (conv5; recycled)
    float* D1     = take((size_t)BB * DCH * MM);     //  67 MB
    float* wpad1  = take((size_t)64 * 16);
    float* wpadG1 = take((size_t)512 * DCH);
    float* wpadH1 = take((size_t)512 * DCH);
    float* aArr   = take(1024);
    float* dArr   = take(1024);
    float* hmaxT  = take(1024 * BB);
    float* fc1T   = take(512 * BB);
    float* codeT  = take(512 * BB);
    // decoder ping-pong inside `big` (2 x 66 MB <= 134 MB)
    float* decA = big;
    float* decB = big + (size_t)BB * 512 * MM;

    auto gemm_cfg = [](int TN, int O, int Bz, dim3& grid, dim3& blk) {
        const int nOt32 = (O + 31) / 32;
        const int wy = nOt32 < 4 ? nOt32 : 4;
        grid = dim3((unsigned)((TN + 31) / 32), (unsigned)((nOt32 + wy - 1) / wy), (unsigned)Bz);
        blk  = dim3(32, (unsigned)wy);
    };
    auto ceil256 = [](size_t n) { return (unsigned)((n + 255) / 256); };
    dim3 gr, bl;

    // weight padding (K-dims not multiple of 8)
    fn_pad_w<<<ceil256((size_t)64 * 16), 256, 0, stream>>>(c1w, wpad1, 64, 12, 16);
    fn_pad_w<<<ceil256((size_t)512 * DCH), 256, 0, stream>>>(g1w, wpadG1, 512, 514, DCH);
    fn_pad_w<<<ceil256((size_t)512 * DCH), 256, 0, stream>>>(h1w, wpadH1, 512, 515, DCH);

    // ---------------- encoder ----------------
    fn_concat16<<<ceil256((size_t)BB * 16 * NN), 256, 0, stream>>>(x, Cov, cat16);

    gemm_cfg(NN, 64, BB, gr, bl);
    fn_gemm_wmma<NN><<<gr, bl, 0, stream>>>(cat16, wpad1, c1b, t64a, 16, 64, 0);
    fn_bn_stats<<<64, 256, 0, stream>>>(t64a, bn1g, bn1b, aArr, dArr, 64, NN, BB);
    fn_bn_apply_relu<<<ceil256((size_t)BB * 64 * NN), 256, 0, stream>>>(t64a, aArr, dArr, 64, NN, BB * 64 * NN);

    fn_gemm_wmma<NN><<<gr, bl, 0, stream>>>(t64a, c2w, c2b, t64b, 64, 64, 0);
    fn_bn_stats<<<64, 256, 0, stream>>>(t64b, bn2g, bn2b, aArr, dArr, 64, NN, BB);
    fn_bn_apply_relu<<<ceil256((size_t)BB * 64 * NN), 256, 0, stream>>>(t64b, aArr, dArr, 64, NN, BB * 64 * NN);

    fn_gemm_wmma<NN><<<gr, bl, 0, stream>>>(t64b, c3w, c3b, t64a, 64, 64, 0);
    fn_bn_stats<<<64, 256, 0, stream>>>(t64a, bn3g, bn3b, aArr, dArr, 64, NN, BB);
    fn_bn_apply_relu<<<ceil256((size_t)BB * 64 * NN), 256, 0, stream>>>(t64a, aArr, dArr, 64, NN, BB * 64 * NN);

    fn_graph_pool<<<ceil256((size_t)BB * NN * 4), 256, 0, stream>>>(t64a, nb, t64b, 64);

    gemm_cfg(NN, 128, BB, gr, bl);
    fn_gemm_wmma<NN><<<gr, bl, 0, stream>>>(t64b, c4w, c4b, t128a, 64, 128, 0);
    fn_bn_stats<<<128, 256, 0, stream>>>(t128a, bn4g, bn4b, aArr, dArr, 128, NN, BB);
    fn_bn_apply_relu<<<ceil256((size_t)BB * 128 * NN), 256, 0, stream>>>(t128a, aArr, dArr, 128, NN, BB * 128 * NN);

    fn_graph_pool<<<ceil256((size_t)BB * NN * 8), 256, 0, stream>>>(t128a, nb, t128b, 128);

    gemm_cfg(NN, 1024, BB, gr, bl);
    fn_gemm_wmma<NN><<<gr, bl, 0, stream>>>(t128b, c5w, c5b, big, 128, 1024, 0);
    fn_bn_stats<<<1024, 256, 0, stream>>>(big, bn5g, bn5b, aArr, dArr, 1024, NN, BB);
    fn_maxpool_bn<<<BB * 1024, 256, 0, stream>>>(big, aArr, dArr, hmaxT, 1024, NN);  // -> [1024][16]

    // FCs on transposed activations: X layout [C][Nn=16], Bz=1
    gemm_cfg(BB, 512, 1, gr, bl);
    fn_gemm_wmma<BB><<<gr, bl, 0, stream>>>(hmaxT, f1w, f1b, fc1T, 1024, 512, 0);
    fn_bn_fc_relu<<<2, 256, 0, stream>>>(fc1T, bn6g, bn6b, 512);
    fn_gemm_wmma<BB><<<gr, bl, 0, stream>>>(fc1T, f2w, f2b, codeT, 512, 512, 0);
    fn_round_code<<<ceil256(512 * BB), 256, 0, stream>>>(codeT, codeP);

    // ---------------- decoder ----------------
    fn_build_dec_in<<<ceil256((size_t)BB * DCH * MM), 256, 0, stream>>>(codeT, D1);

    gemm_cfg(MM, 512, BB, gr, bl);
    fn_gemm_wmma<MM><<<gr, bl, 0, stream>>>(D1,   wpadG1, g1b, decA, DCH, 512, 1);
    fn_gemm_wmma<MM><<<gr, bl, 0, stream>>>(decA, g2w,    g2b, decB, 512, 512, 1);
    gemm_cfg(MM, 3, BB, gr, bl);
    fn_gemm_wmma<MM><<<gr, bl, 0, stream>>>(decB, g3w,    g3b, p1P,  512,   3, 0);

    fn_patch_p1<<<ceil256((size_t)BB * 3 * MM), 256, 0, stream>>>(p1P, D1);

    gemm_cfg(MM, 512, BB, gr, bl);
    fn_gemm_wmma<MM><<<gr, bl, 0, stream>>>(D1,   wpadH1, h1b, decA, DCH, 512, 1);
    fn_gemm_wmma<MM><<<gr, bl, 0, stream>>>(decA, h2w,    h2b, decB, 512, 512, 1);
    gemm_cfg(MM, 3, BB, gr, bl);
    fn_gemm_wmma<MM><<<gr, bl, 0, stream>>>(decB, h3w,    h3b, outP, 512,   3, 0);
}